// LightGCN_17746804867105
// MI455X (gfx1250) — compile-verified
//
#include <hip/hip_runtime.h>

#define VEC_DIM  64
#define THREADS  256
#define WPB      (THREADS / 32)   // waves per block (wave32)
#define EPW      64               // edges per wave
#define PDEPTH   8                // async gather pipeline depth (8 x 256B in flight)

// out = 0.25f * E  (vectorized float4 stream; E stays resident in L2 for gathers)
__global__ void lightgcn_init(const float* __restrict__ E,
                              float* __restrict__ out, int n4) {
  int i = blockIdx.x * blockDim.x + threadIdx.x;
  if (i < n4) {
    float4 v = reinterpret_cast<const float4*>(E)[i];
    float4 r;
    r.x = 0.25f * v.x; r.y = 0.25f * v.y;
    r.z = 0.25f * v.z; r.w = 0.25f * v.w;
    reinterpret_cast<float4*>(out)[i] = r;
  }
}

// Edge-parallel SpMM with a CDNA5 async-gather pipeline:
//   out[row] += 0.75 * val * E[col]
// One wave32 per EPW consecutive edges; lane covers dims [2*lane, 2*lane+1].
// The gathered 256B embedding row for edge k+PDEPTH is DMA'd into LDS via
// global_load_async_to_lds_b64 (tracked by ASYNCcnt) while edge k is consumed
// from LDS (ds_load_b64). Consecutive same-row edges (first half of the edge
// list is row-sorted, avg run ~20) accumulate in registers; a flush is
// 2x global_atomic_add_f32 per lane, resolved in the 192MB L2 where the whole
// 124MB working set (E + out + edge lists) is resident.
__global__ __launch_bounds__(THREADS)
void lightgcn_spmm(const float* __restrict__ E,
                   const float* __restrict__ vals,
                   const int*   __restrict__ rows,
                   const int*   __restrict__ cols,
                   float* __restrict__ out, int nnz) {
  __shared__ float2 stage[WPB][PDEPTH][32];   // 16 KB / block (320 KB per WGP)

  const int  lane = threadIdx.x & 31;         // wave32
  const int  wib  = threadIdx.x >> 5;
  const long wave = (long)blockIdx.x * WPB + wib;
  const long base = wave * (long)EPW;
  if (base >= (long)nnz) return;
  long rem = (long)nnz - base;
  const int n = (int)(rem < EPW ? rem : EPW);
  const long last = (long)nnz - 1;

  // Wave-relative LDS byte address of this lane's slot in chunk 0.
  const unsigned ldsBase = (unsigned)(size_t)(const void*)(&stage[wib][0][lane]);

  // --- prologue: issue PDEPTH async gathers (tail indices clamped) ---
#pragma unroll
  for (int k = 0; k < PDEPTH; ++k) {
    long e = base + k; if (e > last) e = last;
    const int      c    = __builtin_nontemporal_load(&cols[e]);
    const int      gOff = c * (VEC_DIM * 4) + lane * 8;           // byte offset vs E
    const unsigned l    = ldsBase + (unsigned)k * 256u;           // chunk slot
    asm volatile("global_load_async_to_lds_b64 %0, %1, %2"
                 :: "v"(l), "v"(gOff), "s"(E) : "memory");
  }

  int   curRow = __builtin_nontemporal_load(&rows[base]);
  float accx = 0.0f, accy = 0.0f;

  for (int k = 0; k < EPW; ++k) {
    // Exactly PDEPTH+k chunks issued so far; async loads complete in order,
    // so <=PDEPTH-1 outstanding  =>  chunk k has landed in LDS.
    asm volatile("s_wait_asynccnt %0" :: "n"(PDEPTH - 1) : "memory");

    if (k < n) {                               // wave-uniform predicate
      const long  e = base + k;
      const int   r = __builtin_nontemporal_load(&rows[e]);
      const float w = 0.75f * __builtin_nontemporal_load(&vals[e]);

      if (r != curRow) {                       // wave-uniform branch
        float* o = out + (long)curRow * VEC_DIM + lane * 2;
        unsafeAtomicAdd(o + 0, accx);          // global_atomic_add_f32
        unsafeAtomicAdd(o + 1, accy);
        accx = 0.0f; accy = 0.0f;
        curRow = r;
      }
      const float2 v = stage[wib][k & (PDEPTH - 1)][lane];   // ds_load_b64
      accx = fmaf(w, v.x, accx);
      accy = fmaf(w, v.y, accy);
    }

    // WAR guard: our ds_load of this slot must retire before the async DMA
    // below overwrites it (slot (k+PDEPTH)%PDEPTH == slot k%PDEPTH).
    asm volatile("s_wait_dscnt 0" ::: "memory");

    // Steady-state: always issue exactly one chunk (clamped at the tail) so
    // the s_wait_asynccnt immediate above stays valid for every k.
    {
      const int kn = k + PDEPTH;
      long e = base + kn; if (e > last) e = last;
      const int      c    = __builtin_nontemporal_load(&cols[e]);
      const int      gOff = c * (VEC_DIM * 4) + lane * 8;
      const unsigned l    = ldsBase + (unsigned)(kn & (PDEPTH - 1)) * 256u;
      asm volatile("global_load_async_to_lds_b64 %0, %1, %2"
                   :: "v"(l), "v"(gOff), "s"(E) : "memory");
    }
  }

  float* o = out + (long)curRow * VEC_DIM + lane * 2;
  unsafeAtomicAdd(o + 0, accx);
  unsafeAtomicAdd(o + 1, accy);
  // Leftover async ops drain at s_endpgm (implicit wait-idle).
}

// Inputs (setup_inputs order):
//   d_in[0] embed_weights : float32 [150000*64]
//   d_in[1] vals          : float32 [nnz]
//   d_in[2] user_idxs     : int64   [8192]   (UNUSED by the reference output)
//   d_in[3] rows          : int32   [nnz]
//   d_in[4] cols          : int32   [nnz]
// Output: mean_embed flat = (user_embed ++ item_embed) = 150000*64 float32.
// (Reference quirk: every "layer" re-reads the ORIGINAL embeddings, so
//  out = 0.25*E + 0.75*(A @ E) — exactly one SpMM.)
extern "C" void kernel_launch(void* const* d_in, const int* in_sizes, int n_in,
                              void* d_out, int out_size, void* d_ws, size_t ws_size,
                              hipStream_t stream) {
  const float* E    = (const float*)d_in[0];
  const float* vals = (const float*)d_in[1];
  const int*   rows = (const int*)d_in[3];
  const int*   cols = (const int*)d_in[4];
  float*       out  = (float*)d_out;
  const int    nnz  = in_sizes[1];

  // Epilogue-folded init: out = 0.25 * E
  const int n4 = out_size / 4;   // 9.6M floats -> 2.4M float4
  lightgcn_init<<<(n4 + THREADS - 1) / THREADS, THREADS, 0, stream>>>(E, out, n4);

  // SpMM: out += 0.75 * A @ E  (one wave per EPW consecutive edges)
  const long waves  = ((long)nnz + EPW - 1) / EPW;
  const int  blocks = (int)((waves + WPB - 1) / WPB);
  lightgcn_spmm<<<blocks, THREADS, 0, stream>>>(E, vals, rows, cols, out, nnz);
}